// Rewirescorelayer_63024350102001
// MI455X (gfx1250) — compile-verified
//
#include <hip/hip_runtime.h>
#include <hip/hip_bf16.h>

typedef __attribute__((ext_vector_type(2))) float v2f;
typedef __attribute__((ext_vector_type(8))) float v8f;

#define IN_F 256
#define HD   256     // H*D
#define NH   4
#define DH   64
#define WIN  128
#define HALFW 64
#define TOPK 32

// ---------------------------------------------------------------------------
// Kernel 1: zero the NxN output (harness poisons it with 0xAA)
// ---------------------------------------------------------------------------
__global__ void zero_out_kernel(float4* __restrict__ p, size_t n4) {
    size_t idx = (size_t)blockIdx.x * blockDim.x + threadIdx.x;
    size_t stride = (size_t)gridDim.x * blockDim.x;
    float4 z = make_float4(0.f, 0.f, 0.f, 0.f);
    for (size_t i = idx; i < n4; i += stride) p[i] = z;
}

// ---------------------------------------------------------------------------
// Kernel 2: fused Q/K projection GEMM, fp32 WMMA 16x16x4.
//   C[n,256] = X[n,256] @ W[256,256] + b
//   One wave computes one 16x16 C tile, K-loop in steps of 4.
//   A 16x4 layout (ISA 7.12.2): lane r=lane&15, hl=lane>>4 ; a.x = X[r, k+2*hl],
//   a.y = X[r, k+2*hl+1].  B 4x16: b.x = W[k+2*hl, col], b.y = W[k+2*hl+1, col].
//   C/D: vgpr v -> M = v + 8*hl, N = r.
// ---------------------------------------------------------------------------
__global__ void qk_gemm_kernel(const float* __restrict__ X,
                               const float* __restrict__ Wq,
                               const float* __restrict__ bq,
                               const float* __restrict__ Wk,
                               const float* __restrict__ bk,
                               float* __restrict__ Qo,
                               float* __restrict__ Ko,
                               int n) {
    const int wave = threadIdx.x >> 5;
    const int lane = threadIdx.x & 31;
    const int r    = lane & 15;
    const int hl   = lane >> 4;          // 0 or 1

    const int row0 = blockIdx.x * 16;
    const int col0 = (blockIdx.y * 4 + wave) * 16;
    if (row0 >= n) return;

    const float* W    = (blockIdx.z == 0) ? Wq : Wk;
    const float* bias = (blockIdx.z == 0) ? bq : bk;
    float*       O    = (blockIdx.z == 0) ? Qo : Ko;

    const float* xrow = X + (size_t)(row0 + r) * IN_F;
    const int col = col0 + r;

    v8f acc = {};
    for (int k = 0; k < IN_F; k += 4) {
        const int ka = k + hl * 2;
        v2f a, b;
        a.x = xrow[ka];
        a.y = xrow[ka + 1];
        b.x = W[(size_t)ka       * HD + col];
        b.y = W[(size_t)(ka + 1) * HD + col];
        acc = __builtin_amdgcn_wmma_f32_16x16x4_f32(
            /*neg_a=*/false, a, /*neg_b=*/false, b,
            /*c_mod=*/(short)0, acc, /*reuse_a=*/false, /*reuse_b=*/false);
    }

    const float bv = bias[col];
#pragma unroll
    for (int v = 0; v < 8; ++v) {
        const int row = row0 + v + hl * 8;
        O[(size_t)row * HD + col] = acc[v] + bv;
    }
}

// ---------------------------------------------------------------------------
// Kernel 3: per-node windowed attention + stable top-32 scatter.
//   block = 128 threads (one per window slot), one block per node.
// ---------------------------------------------------------------------------
__global__ void attn_topk_kernel(const float* __restrict__ Q,
                                 const float* __restrict__ K,
                                 const int* __restrict__ nn_per_graph,
                                 int n_graphs,
                                 const int* __restrict__ num_rel_p,
                                 float* __restrict__ out,
                                 int n) {
    __shared__ float qs[HD];          // this node's Q row (4 heads x 64)
    __shared__ float sc[NH][WIN];     // per-head scores, then exp()s
    __shared__ float mx[NH];
    __shared__ float sm[NH];
    __shared__ float sv[WIN];         // head-averaged attn (invalid -> -1)
    __shared__ int   seg[2];          // start, end (absolute column range)

    const int i   = blockIdx.x;
    const int tid = threadIdx.x;      // window slot m

    if (tid == 0) {
        // segment = tile(num_nodes_per_graph, num_relation); cumsum; searchsorted
        const int nrel = num_rel_p[0];
        const long G   = (long)n_graphs * (long)nrel;
        long c = 0;
        int s = 0, e = 0;
        bool found = false;
        for (long g = 0; g < G; ++g) {
            const int len = nn_per_graph[g % n_graphs];
            const long c2 = c + (long)len;
            if ((long)i < c2) { s = (int)c; e = (int)c2; found = true; break; }
            c = c2;
        }
        if (!found) { s = (int)c; e = (int)c; }   // pos == G edge case
        int start = s; if (i - HALFW > start) start = i - HALFW;
        int end   = e; if (i + HALFW < end)   end   = i + HALFW;
        seg[0] = start;
        seg[1] = end;
    }

    // load Q row into LDS (256 floats / 128 threads)
    qs[tid]       = Q[(size_t)i * HD + tid];
    qs[tid + WIN] = Q[(size_t)i * HD + tid + WIN];
    __syncthreads();

    const int start = seg[0];
    const int end   = seg[1];
    const int j     = start + tid;
    const bool valid = (j < end);

    // scores for 4 heads (reference: QK / sqrt(D) / tau = QK * 0.25)
    float s0 = 0.f, s1 = 0.f, s2 = 0.f, s3 = 0.f;
    if (valid) {
        const float4* kr = reinterpret_cast<const float4*>(K + (size_t)j * HD);
#pragma unroll 4
        for (int d4 = 0; d4 < DH / 4; ++d4) {
            {
                float4 kv = kr[d4];
                const float* q = qs + 0 * DH + d4 * 4;
                s0 += q[0] * kv.x + q[1] * kv.y + q[2] * kv.z + q[3] * kv.w;
            }
            {
                float4 kv = kr[DH / 4 + d4];
                const float* q = qs + 1 * DH + d4 * 4;
                s1 += q[0] * kv.x + q[1] * kv.y + q[2] * kv.z + q[3] * kv.w;
            }
            {
                float4 kv = kr[2 * DH / 4 + d4];
                const float* q = qs + 2 * DH + d4 * 4;
                s2 += q[0] * kv.x + q[1] * kv.y + q[2] * kv.z + q[3] * kv.w;
            }
            {
                float4 kv = kr[3 * DH / 4 + d4];
                const float* q = qs + 3 * DH + d4 * 4;
                s3 += q[0] * kv.x + q[1] * kv.y + q[2] * kv.z + q[3] * kv.w;
            }
        }
        s0 *= 0.25f; s1 *= 0.25f; s2 *= 0.25f; s3 *= 0.25f;
    }
    // invalid slots: K zero-padded in reference => score exactly 0, still in softmax
    sc[0][tid] = s0; sc[1][tid] = s1; sc[2][tid] = s2; sc[3][tid] = s3;
    __syncthreads();

    if (tid < NH) {
        float m = sc[tid][0];
        for (int t = 1; t < WIN; ++t) { float v = sc[tid][t]; if (v > m) m = v; }
        mx[tid] = m;
    }
    __syncthreads();

#pragma unroll
    for (int h = 0; h < NH; ++h) sc[h][tid] = __expf(sc[h][tid] - mx[h]);
    __syncthreads();

    if (tid < NH) {
        float s = 0.f;
        for (int t = 0; t < WIN; ++t) s += sc[tid][t];
        sm[tid] = s;
    }
    __syncthreads();

    const float attn = 0.25f * (sc[0][tid] / sm[0] + sc[1][tid] / sm[1] +
                                sc[2][tid] / sm[2] + sc[3][tid] / sm[3]);
    sv[tid] = valid ? attn : -1.0f;   // invalid can never reach top-32
    __syncthreads();

    // stable top-32: rank = #{ m' : v' > v  ||  (v' == v && m' < tid) }
    const float v = sv[tid];
    int cnt = 0;
    for (int t = 0; t < WIN; ++t) {
        const float u = sv[t];
        cnt += (u > v) || (u == v && t < tid);
    }
    if (valid && cnt < TOPK) {
        out[(size_t)i * n + j] = 1.0f;   // y_hard one-hot; y == y_hard numerically
    }
}

// ---------------------------------------------------------------------------
extern "C" void kernel_launch(void* const* d_in, const int* in_sizes, int n_in,
                              void* d_out, int out_size, void* d_ws, size_t ws_size,
                              hipStream_t stream) {
    const float* X  = (const float*)d_in[0];   // [n, 256]
    const float* Wq = (const float*)d_in[1];   // [256, 256]
    const float* bq = (const float*)d_in[2];   // [256]
    const float* Wk = (const float*)d_in[3];   // [256, 256]
    const float* bk = (const float*)d_in[4];   // [256]
    const int* nn_per_graph = (const int*)d_in[5];   // [n_graphs]
    const int* num_rel      = (const int*)d_in[6];   // scalar

    const int n        = in_sizes[0] / IN_F;
    const int n_graphs = in_sizes[5];
    float* out = (float*)d_out;

    float* Qw = (float*)d_ws;                       // n*256 floats
    float* Kw = Qw + (size_t)n * HD;                // n*256 floats

    // 1) zero the NxN output
    const size_t n4 = ((size_t)n * (size_t)n) / 4;
    int zblocks = (int)((n4 + 255) / 256);
    if (zblocks > 16384) zblocks = 16384;
    zero_out_kernel<<<zblocks, 256, 0, stream>>>((float4*)out, n4);

    // 2) Q/K projections via fp32 WMMA
    dim3 ggrid(n / 16, HD / 64, 2);                 // row tiles x (4 col-tiles/block) x {Q,K}
    qk_gemm_kernel<<<ggrid, 128, 0, stream>>>(X, Wq, bq, Wk, bk, Qw, Kw, n);

    // 3) windowed attention + top-32 scatter
    attn_topk_kernel<<<n, WIN, 0, stream>>>(Qw, Kw, nn_per_graph, n_graphs,
                                            num_rel, out, n);
}